// GCNModelVAE_28905129902430
// MI455X (gfx1250) — compile-verified
//
#include <hip/hip_runtime.h>
#include <hip/hip_bf16.h>

// ---------------------------------------------------------------------------
// GCN-VAE forward for MI455X (gfx1250, wave32, WMMA f32 16x16x4).
//
// Sizes (compile-time, match reference):
//   N = 16384 nodes, F_IN = 512, E = 524288 edges, H1 = 32, H2 = 16
// Output: adj [N*N] ++ mu [N*16] ++ logvar [N*16]  (f32)
// ---------------------------------------------------------------------------

#define NN     16384
#define FIN    512
#define EDGES  524288
#define H1F    32
#define H2F    16

typedef __attribute__((ext_vector_type(2))) float v2f;
typedef __attribute__((ext_vector_type(8))) float v8f;

// ---------------- graph preprocessing ----------------

__global__ void k_init(int* cnt, int* cursor) {
    int i = blockIdx.x * 256 + threadIdx.x;
    cnt[i] = 0;
    cursor[i] = 0;
}

__global__ void k_count(const long long* __restrict__ dst, int* __restrict__ cnt) {
    int e = blockIdx.x * 256 + threadIdx.x;
    atomicAdd(&cnt[(int)dst[e]], 1);
}

__global__ void k_dinv(const int* __restrict__ cnt, float* __restrict__ dinv) {
    int i = blockIdx.x * 256 + threadIdx.x;
    // +1 for the self loop -> deg >= 1
    dinv[i] = rsqrtf((float)(cnt[i] + 1));
}

// exclusive scan of 16384 ints, single block of 1024 threads (16 per thread)
__global__ void k_scan(const int* __restrict__ cnt, int* __restrict__ row_start) {
    __shared__ int part[1024];
    int t = threadIdx.x;
    int base = t * 16;
    int loc[16];
    int s = 0;
#pragma unroll
    for (int k = 0; k < 16; ++k) { loc[k] = s; s += cnt[base + k]; }
    part[t] = s;
    __syncthreads();
    for (int off = 1; off < 1024; off <<= 1) {
        int v = (t >= off) ? part[t - off] : 0;
        __syncthreads();
        part[t] += v;
        __syncthreads();
    }
    int pre = (t == 0) ? 0 : part[t - 1];
#pragma unroll
    for (int k = 0; k < 16; ++k) row_start[base + k] = pre + loc[k];
    if (t == 1023) row_start[NN] = pre + s;
}

__global__ void k_fill(const long long* __restrict__ src, const long long* __restrict__ dst,
                       const int* __restrict__ row_start, int* __restrict__ cursor,
                       int* __restrict__ csr_src) {
    int e = blockIdx.x * 256 + threadIdx.x;
    int s = (int)src[e];
    int d = (int)dst[e];
    int p = atomicAdd(&cursor[d], 1);
    csr_src[row_start[d] + p] = s;
}

// ---------------- layer 1: xs = dinv * (x @ W1)  [N,32] ----------------
// 4 waves / block, each wave owns one 16-row i-tile; two 16-wide n-tiles share A.

__global__ void k_xw1(const float* __restrict__ x, const float* __restrict__ W1,
                      const float* __restrict__ dinv, float* __restrict__ xs) {
    int wave = threadIdx.x >> 5;
    int lane = threadIdx.x & 31;
    int l = lane & 15, h = lane >> 4;
    int i0 = (blockIdx.x * 4 + wave) * 16;

    const float* xrow = x + (size_t)(i0 + l) * FIN;
    v8f c0 = {}, c1 = {};
    for (int k0 = 0; k0 < FIN; k0 += 4) {
        v2f a = *(const v2f*)(xrow + k0 + 2 * h);           // A: K = 2h, 2h+1
        v2f b0, b1;                                          // B: v0={K0|K2}, v1={K1|K3}
        b0.x = W1[(k0 + 2 * h) * H1F + l];
        b0.y = W1[(k0 + 1 + 2 * h) * H1F + l];
        b1.x = W1[(k0 + 2 * h) * H1F + 16 + l];
        b1.y = W1[(k0 + 1 + 2 * h) * H1F + 16 + l];
        c0 = __builtin_amdgcn_wmma_f32_16x16x4_f32(false, a, false, b0, (short)0, c0, false, false);
        c1 = __builtin_amdgcn_wmma_f32_16x16x4_f32(false, a, false, b1, (short)0, c1, false, false);
    }
#pragma unroll
    for (int r = 0; r < 8; ++r) {
        int m = r + 8 * h;
        float dv = dinv[i0 + m];
        xs[(size_t)(i0 + m) * H1F + l]      = dv * c0[r];
        xs[(size_t)(i0 + m) * H1F + 16 + l] = dv * c1[r];
    }
}

// ---------------- gather: h1 = dinv*(sum_nbr xs + xs_self) + b1 ----------------
// one wave per node, lane = feature (32 feats). CSR rows are 128B coalesced.

__global__ void k_gather32(const float* __restrict__ xs, const int* __restrict__ row_start,
                           const int* __restrict__ csr_src, const float* __restrict__ dinv,
                           const float* __restrict__ b1, float* __restrict__ h1) {
    int d = blockIdx.x * 8 + (threadIdx.x >> 5);
    int lane = threadIdx.x & 31;
    int s0 = row_start[d], s1 = row_start[d + 1];
    float acc = xs[(size_t)d * H1F + lane];   // self loop contribution
    for (int j = s0; j < s1; ++j) {
        int s = csr_src[j];
        acc += xs[(size_t)s * H1F + lane];
    }
    h1[(size_t)d * H1F + lane] = dinv[d] * acc + b1[lane];
}

// ---------------- layers 2+3 fused: ys = dinv * (h1 @ [W2 | W3])  [N,32] ----------------

__global__ void k_y23(const float* __restrict__ h1, const float* __restrict__ W2,
                      const float* __restrict__ W3, const float* __restrict__ dinv,
                      float* __restrict__ ys) {
    int wave = threadIdx.x >> 5;
    int lane = threadIdx.x & 31;
    int l = lane & 15, h = lane >> 4;
    int i0 = (blockIdx.x * 4 + wave) * 16;

    const float* arow = h1 + (size_t)(i0 + l) * H1F;
    v8f c0 = {}, c1 = {};
#pragma unroll
    for (int k0 = 0; k0 < H1F; k0 += 4) {
        v2f a = *(const v2f*)(arow + k0 + 2 * h);
        v2f b0, b1;
        b0.x = W2[(k0 + 2 * h) * H2F + l];
        b0.y = W2[(k0 + 1 + 2 * h) * H2F + l];
        b1.x = W3[(k0 + 2 * h) * H2F + l];
        b1.y = W3[(k0 + 1 + 2 * h) * H2F + l];
        c0 = __builtin_amdgcn_wmma_f32_16x16x4_f32(false, a, false, b0, (short)0, c0, false, false);
        c1 = __builtin_amdgcn_wmma_f32_16x16x4_f32(false, a, false, b1, (short)0, c1, false, false);
    }
#pragma unroll
    for (int r = 0; r < 8; ++r) {
        int m = r + 8 * h;
        float dv = dinv[i0 + m];
        ys[(size_t)(i0 + m) * 32 + l]      = dv * c0[r];   // mu half
        ys[(size_t)(i0 + m) * 32 + 16 + l] = dv * c1[r];   // logvar half
    }
}

// ---------------- gather: mu / logvar (lanes 0-15 / 16-31) ----------------

__global__ void k_gather_out(const float* __restrict__ ys, const int* __restrict__ row_start,
                             const int* __restrict__ csr_src, const float* __restrict__ dinv,
                             const float* __restrict__ b2, const float* __restrict__ b3,
                             float* __restrict__ mu_out, float* __restrict__ lv_out,
                             float* __restrict__ muw) {
    int d = blockIdx.x * 8 + (threadIdx.x >> 5);
    int lane = threadIdx.x & 31;
    int s0 = row_start[d], s1 = row_start[d + 1];
    float acc = ys[(size_t)d * 32 + lane];
    for (int j = s0; j < s1; ++j) {
        int s = csr_src[j];
        acc += ys[(size_t)s * 32 + lane];
    }
    float bias = (lane < 16) ? b2[lane] : b3[lane - 16];
    float v = dinv[d] * acc + bias;
    if (lane < 16) {
        mu_out[(size_t)d * H2F + lane] = v;
        muw[(size_t)d * H2F + lane] = v;      // L2-resident copy for the decoder
    } else {
        lv_out[(size_t)d * H2F + (lane - 16)] = v;
    }
}

// ---------------- decode: adj = sigmoid(mu @ mu.T), WMMA f32 16x16x4 ----------------
// 4 waves/block, wave owns i-tile; loops 16 j-tiles per grid.y -> streams a 16x256 strip.
// B-tile of mu@mu.T has the same per-lane layout as the A-tile (transpose symmetry).

__global__ void k_decode(const float* __restrict__ mu, float* __restrict__ adj) {
    int wave = threadIdx.x >> 5;
    int lane = threadIdx.x & 31;
    int l = lane & 15, h = lane >> 4;
    int i0 = (blockIdx.x * 4 + wave) * 16;

    v2f a[4];
    const float* arow = mu + (size_t)(i0 + l) * H2F;
#pragma unroll
    for (int q = 0; q < 4; ++q) a[q] = *(const v2f*)(arow + 4 * q + 2 * h);

    int jt0 = blockIdx.y * 16;
    for (int t = 0; t < 16; ++t) {
        int j0 = (jt0 + t) * 16;
        const float* brow = mu + (size_t)(j0 + l) * H2F;
        v2f b[4];
#pragma unroll
        for (int q = 0; q < 4; ++q) b[q] = *(const v2f*)(brow + 4 * q + 2 * h);
        v8f c = {};
#pragma unroll
        for (int q = 0; q < 4; ++q)
            c = __builtin_amdgcn_wmma_f32_16x16x4_f32(false, a[q], false, b[q], (short)0, c, false, false);
#pragma unroll
        for (int r = 0; r < 8; ++r) {
            float v = c[r];
            float sg = __builtin_amdgcn_rcpf(1.0f + __expf(-v));   // v_exp_f32 + v_rcp_f32
            adj[(size_t)(i0 + r + 8 * h) * NN + j0 + l] = sg;
        }
    }
}

// ---------------- host launcher ----------------

extern "C" void kernel_launch(void* const* d_in, const int* in_sizes, int n_in,
                              void* d_out, int out_size, void* d_ws, size_t ws_size,
                              hipStream_t stream) {
    (void)in_sizes; (void)n_in; (void)out_size; (void)ws_size;

    const float*     x   = (const float*)d_in[0];
    const long long* ei  = (const long long*)d_in[1];   // int64 [2, E]
    const float*     W1  = (const float*)d_in[2];
    const float*     b1  = (const float*)d_in[3];
    const float*     W2  = (const float*)d_in[4];
    const float*     b2  = (const float*)d_in[5];
    const float*     W3  = (const float*)d_in[6];
    const float*     b3  = (const float*)d_in[7];

    const long long* src = ei;
    const long long* dst = ei + EDGES;

    float* out    = (float*)d_out;
    float* adj    = out;
    float* mu_out = out + (size_t)NN * NN;
    float* lv_out = mu_out + (size_t)NN * H2F;

    // workspace carve-up (all offsets 256B aligned)
    char* w = (char*)d_ws;
    int*   cnt       = (int*)(w + 0);               //  64 KB
    int*   cursor    = (int*)(w + 65536);           //  64 KB
    int*   row_start = (int*)(w + 131072);          //  64 KB + 4 (N+1 ints)
    float* dinv      = (float*)(w + 197120);        //  64 KB
    int*   csr_src   = (int*)(w + 262656);          //   2 MB
    float* xs        = (float*)(w + 2359808);       //   2 MB  (N x 32)
    float* h1        = (float*)(w + 4456960);       //   2 MB  (N x 32)
    float* ys        = (float*)(w + 6554112);       //   2 MB  (N x 32)
    float* muw       = (float*)(w + 8651264);       //   1 MB  (N x 16)

    // graph prep
    k_init <<<NN / 256, 256, 0, stream>>>(cnt, cursor);
    k_count<<<EDGES / 256, 256, 0, stream>>>(dst, cnt);
    k_dinv <<<NN / 256, 256, 0, stream>>>(cnt, dinv);
    k_scan <<<1, 1024, 0, stream>>>(cnt, row_start);
    k_fill <<<EDGES / 256, 256, 0, stream>>>(src, dst, row_start, cursor, csr_src);

    // GCN layer 1
    k_xw1     <<<NN / 64, 128, 0, stream>>>(x, W1, dinv, xs);
    k_gather32<<<NN / 8, 256, 0, stream>>>(xs, row_start, csr_src, dinv, b1, h1);

    // GCN layers 2 + 3 (fused)
    k_y23       <<<NN / 64, 128, 0, stream>>>(h1, W2, W3, dinv, ys);
    k_gather_out<<<NN / 8, 256, 0, stream>>>(ys, row_start, csr_src, dinv, b2, b3,
                                             mu_out, lv_out, muw);

    // inner-product decoder (dominant cost: 1.07 GB streaming store)
    dim3 dgrid(NN / 64, NN / 256);   // (256, 64): 4 i-tiles/block x 16 j-tiles/wave
    k_decode<<<dgrid, 128, 0, stream>>>(muw, adj);
}